// GeomGCNSingleChannel_84593675862585
// MI455X (gfx1250) — compile-verified
//
#include <hip/hip_runtime.h>
#include <hip/hip_bf16.h>

// Problem constants (match reference)
#define NN   100000
#define EE   1600000
#define DD   9
#define INF  128
#define OUTF 64
#define COLS (DD * OUTF)   // 576
#define MT   (NN / 16)     // 6250 row tiles (exact)
#define CT   (COLS / 16)   // 36 col tiles
#define NTILES (MT * CT)   // 225000, divisible by 8

typedef __attribute__((ext_vector_type(2))) float v2f;
typedef __attribute__((ext_vector_type(8))) float v8f;

// ---------------------------------------------------------------------------
// Kernel 1: Wh[n][d*64+o] = (feature[n,:] . W[d][o][:]) * norm[n]
// One wave per 16x16 output tile, K=128 via 32x V_WMMA_F32_16X16X4_F32.
// W is [D][OUT][IN] row-major, so column c = d*64+o has its K-vector at
// W + c*128 — i.e. B[k][c] = (W + c*128)[k].
// ---------------------------------------------------------------------------
__global__ void __launch_bounds__(256)
geomgcn_wmma_gemm(const float* __restrict__ feat,
                  const float* __restrict__ norm,
                  const float* __restrict__ W,
                  float* __restrict__ Wh) {
  const int lane = threadIdx.x & 31;
  const int wave = threadIdx.x >> 5;          // 0..7
  const int tile = blockIdx.x * 8 + wave;     // 0..NTILES-1
  const int mt = tile / CT;                   // row tile
  const int ct = tile % CT;                   // col tile (a block's 8 tiles may span two mt)
  const int half = lane >> 4;                 // 0: K=4k..4k+1  1: K=4k+2..4k+3
  const int lrow = lane & 15;

  const int nb  = mt * 16;                    // base node of this tile
  const int col = ct * 16 + lrow;             // global output column (= d*64+o)

  const float* arow = feat + (size_t)(nb + lrow) * INF; // A row for this lane
  const float* brow = W    + (size_t)col * INF;         // B column's K-vector

  v8f c = {0.f, 0.f, 0.f, 0.f, 0.f, 0.f, 0.f, 0.f};
#pragma unroll
  for (int kk = 0; kk < INF / 4; ++kk) {
    const int ko = kk * 4 + half * 2;
    v2f a = *(const v2f*)(arow + ko);   // A[lrow][ko], A[lrow][ko+1]
    v2f b = *(const v2f*)(brow + ko);   // B[ko][col], B[ko+1][col]
    // 8 args: (neg_a, A, neg_b, B, c_mod, C, reuse_a, reuse_b)
    c = __builtin_amdgcn_wmma_f32_16x16x4_f32(
        false, a, false, b, (short)0, c, false, false);
  }

  // C layout: VGPR r -> node nb + r + 8*half, column col. Scale by norm[node].
#pragma unroll
  for (int r = 0; r < 8; ++r) {
    const int node = nb + r + 8 * half;
    Wh[(size_t)node * COLS + col] = c[r] * norm[node];
  }
}

// ---------------------------------------------------------------------------
// Kernel 2: zero the accumulation buffer (d_out) — required every call since
// the finalize pass overwrote it last call and the harness doesn't re-poison.
// ---------------------------------------------------------------------------
__global__ void __launch_bounds__(256)
geomgcn_zero(float4* __restrict__ out, long n4) {
  long i = (long)blockIdx.x * blockDim.x + threadIdx.x;
  const long stride = (long)gridDim.x * blockDim.x;
  const float4 z = {0.f, 0.f, 0.f, 0.f};
  for (; i < n4; i += stride) out[i] = z;
}

// ---------------------------------------------------------------------------
// Kernel 3: edge scatter. Thread t handles (edge e = t>>6, column o = t&63):
//   out[dst[e]][ediv[e]*64 + o] += Wh[src[e]][ediv[e]*64 + o]
// 64 consecutive lanes gather 256 B contiguous and atomically add 256 B
// contiguous -> fully coalesced; hardware global_atomic_add_f32.
// ---------------------------------------------------------------------------
__global__ void __launch_bounds__(256)
geomgcn_scatter(const float* __restrict__ Wh,
                const int* __restrict__ src,
                const int* __restrict__ dst,
                const int* __restrict__ ediv,
                float* __restrict__ out) {
  const long t = (long)blockIdx.x * blockDim.x + threadIdx.x; // < E*64
  const int e = (int)(t >> 6);
  const int o = (int)(t & 63);
  const int s  = src[e];
  const int d  = dst[e];
  const int dv = ediv[e];
  const float v = Wh[(size_t)s * COLS + dv * OUTF + o];
  unsafeAtomicAdd(out + (size_t)d * COLS + dv * OUTF + o, v);
}

// ---------------------------------------------------------------------------
// Kernel 4: out = relu(out * norm[n]), in place.
// ---------------------------------------------------------------------------
__global__ void __launch_bounds__(256)
geomgcn_finalize(float* __restrict__ out, const float* __restrict__ norm) {
  const long total = (long)NN * COLS;
  long i = (long)blockIdx.x * blockDim.x + threadIdx.x;
  const long stride = (long)gridDim.x * blockDim.x;
  for (; i < total; i += stride) {
    const int n = (int)(i / COLS);
    const float v = out[i] * norm[n];
    out[i] = v > 0.f ? v : 0.f;
  }
}

extern "C" void kernel_launch(void* const* d_in, const int* in_sizes, int n_in,
                              void* d_out, int out_size, void* d_ws, size_t ws_size,
                              hipStream_t stream) {
  const float* feat = (const float*)d_in[0]; // [N,128]
  const float* norm = (const float*)d_in[1]; // [N,1]
  const float* W    = (const float*)d_in[2]; // [9,64,128]
  const int*   src  = (const int*)d_in[3];   // [E]
  const int*   dst  = (const int*)d_in[4];   // [E]
  const int*   ediv = (const int*)d_in[5];   // [E]
  float* out = (float*)d_out;                // [N, 576]
  float* Wh  = (float*)d_ws;                 // scratch: N*576 floats = 230.4 MB

  // Zero output accumulator (independent of GEMM, launch first).
  const long n4 = (long)NN * COLS / 4;
  geomgcn_zero<<<(unsigned)((n4 + 255) / 256), 256, 0, stream>>>((float4*)out, n4);

  // WMMA GEMM -> Wh.
  geomgcn_wmma_gemm<<<NTILES / 8, 256, 0, stream>>>(feat, norm, W, Wh);

  // Edge scatter-add (E*64 threads).
  const long st = (long)EE * OUTF;
  geomgcn_scatter<<<(unsigned)(st / 256), 256, 0, stream>>>(Wh, src, dst, ediv, out);

  // norm * relu, in place.
  geomgcn_finalize<<<4096, 256, 0, stream>>>(out, norm);
}